// GAT_76759655514229
// MI455X (gfx1250) — compile-verified
//
#include <hip/hip_runtime.h>
#include <math.h>

typedef float v2f __attribute__((ext_vector_type(2)));
typedef float v8f __attribute__((ext_vector_type(8)));

#define NEG_SLOPE 0.2f

// ---------------------------------------------------------------------------
// Utility: fill a float buffer with a constant (used for -inf / 0 init).
// ---------------------------------------------------------------------------
__global__ void gat_fill_kernel(float* __restrict__ p, float v, size_t n) {
  size_t i = (size_t)blockIdx.x * blockDim.x + threadIdx.x;
  if (i < n) p[i] = v;
}

// ---------------------------------------------------------------------------
// Float atomic max via sign-split integer atomics (works with -inf init).
// Positive values: int compare order == float order.
// Negative values: uint compare order is reversed float order -> atomicMin.
// ---------------------------------------------------------------------------
__device__ __forceinline__ void atomicMaxF(float* addr, float val) {
  if (val >= 0.0f) {
    atomicMax((int*)addr, __float_as_int(val));
  } else {
    atomicMin((unsigned int*)addr, __float_as_uint(val));
  }
}

// ---------------------------------------------------------------------------
// fp32 WMMA GEMM: C[M,Ncols] = A[M,K] * B[K,Ncols], row-major, one wave per
// 16x16 output tile, K stepped by 4 with V_WMMA_F32_16X16X4_F32.
//
// VGPR layouts per CDNA5 ISA 7.12.2:
//   A 16x4 f32 : lane(M) = lane&15 ; K = 2*(lane>>4) + vgpr_index
//   B 4x16 f32 : lane(N) = lane&15 ; K = 2*(lane>>4) + vgpr_index
//   C/D 16x16  : vgpr r -> row r + 8*(lane>>4), col = lane&15
// ---------------------------------------------------------------------------
__global__ void gat_wmma_gemm_f32(const float* __restrict__ A,
                                  const float* __restrict__ B,
                                  float* __restrict__ C,
                                  int M, int K, int Ncols) {
  int wid  = (int)(((size_t)blockIdx.x * blockDim.x + threadIdx.x) >> 5);
  int lane = threadIdx.x & 31;
  int tilesN = Ncols >> 4;
  int tm = wid / tilesN;
  int tn = wid - tm * tilesN;
  if (tm * 16 >= M) return;

  int half = lane >> 4;   // selects K pair {0,1} vs {2,3}
  int l16  = lane & 15;

  const float* Arow = A + (size_t)(tm * 16 + l16) * K + 2 * half;
  const float* Bcol = B + (size_t)(2 * half) * Ncols + (tn * 16 + l16);

  v8f acc = {};
#pragma unroll 4
  for (int k = 0; k < K; k += 4) {
    v2f a, b;
    a.x = Arow[k];
    a.y = Arow[k + 1];
    b.x = Bcol[(size_t)k * Ncols];
    b.y = Bcol[(size_t)(k + 1) * Ncols];
    acc = __builtin_amdgcn_wmma_f32_16x16x4_f32(
        /*neg_a=*/false, a, /*neg_b=*/false, b,
        /*c_mod=*/(short)0, acc, /*reuse_a=*/false, /*reuse_b=*/false);
  }

  float* Crow = C + (size_t)(tm * 16 + 8 * half) * Ncols + (tn * 16 + l16);
#pragma unroll
  for (int r = 0; r < 8; ++r) Crow[(size_t)r * Ncols] = acc[r];
}

// ---------------------------------------------------------------------------
// el/er attention scores: one wave per (node, head); D-dim dot product with
// attn_l/attn_r + wave32 shuffle reduction.
// ---------------------------------------------------------------------------
__global__ void gat_attn_scores(const float* __restrict__ feat,
                                const float* __restrict__ attn_l,
                                const float* __restrict__ attn_r,
                                float* __restrict__ el, float* __restrict__ er,
                                int NH, int H, int D) {
  int wave = (int)(((size_t)blockIdx.x * blockDim.x + threadIdx.x) >> 5);
  int lane = threadIdx.x & 31;
  if (wave >= NH) return;
  int h = wave % H;
  const float* f = feat + (size_t)wave * D;
  float sl = 0.0f, sr = 0.0f;
  for (int i = lane; i < D; i += 32) {
    float v = f[i];
    sl += v * attn_l[h * D + i];
    sr += v * attn_r[h * D + i];
  }
#pragma unroll
  for (int o = 16; o > 0; o >>= 1) {
    sl += __shfl_xor(sl, o, 32);
    sr += __shfl_xor(sr, o, 32);
  }
  if (lane == 0) {
    el[wave] = sl;
    er[wave] = sr;
  }
}

// ---------------------------------------------------------------------------
// Edge pass 1: per-dst segment max of leaky-relu(el[src]+er[dst]).
// ---------------------------------------------------------------------------
__global__ void gat_edge_max(const float* __restrict__ el,
                             const float* __restrict__ er,
                             const int* __restrict__ src,
                             const int* __restrict__ dst,
                             float* __restrict__ m, int E, int H) {
  int i = blockIdx.x * blockDim.x + threadIdx.x;
  if (i >= E * H) return;
  int e = i / H, h = i - e * H;
  int s = src[e], d = dst[e];
  float v = el[s * H + h] + er[d * H + h];
  v = v > 0.0f ? v : NEG_SLOPE * v;
  atomicMaxF(&m[d * H + h], v);
}

// ---------------------------------------------------------------------------
// Edge pass 2: denom[dst] += exp(score - m[dst]).
// ---------------------------------------------------------------------------
__global__ void gat_edge_expsum(const float* __restrict__ el,
                                const float* __restrict__ er,
                                const float* __restrict__ m,
                                float* __restrict__ den,
                                const int* __restrict__ src,
                                const int* __restrict__ dst, int E, int H) {
  int i = blockIdx.x * blockDim.x + threadIdx.x;
  if (i >= E * H) return;
  int e = i / H, h = i - e * H;
  int s = src[e], d = dst[e];
  float v = el[s * H + h] + er[d * H + h];
  v = v > 0.0f ? v : NEG_SLOPE * v;
  atomicAdd(&den[d * H + h], expf(v - m[d * H + h]));
}

// ---------------------------------------------------------------------------
// Edge pass 3: out[dst,h,:] += alpha * feat[src,h,:]; one wave per (edge,head).
// ---------------------------------------------------------------------------
__global__ void gat_aggregate(const float* __restrict__ feat,
                              const float* __restrict__ el,
                              const float* __restrict__ er,
                              const float* __restrict__ m,
                              const float* __restrict__ den,
                              const int* __restrict__ src,
                              const int* __restrict__ dst,
                              float* __restrict__ out, int E, int H, int D) {
  int wave = (int)(((size_t)blockIdx.x * blockDim.x + threadIdx.x) >> 5);
  int lane = threadIdx.x & 31;
  if (wave >= E * H) return;
  int e = wave / H, h = wave - e * H;
  int s = src[e], d = dst[e];
  float v = el[s * H + h] + er[d * H + h];
  v = v > 0.0f ? v : NEG_SLOPE * v;
  float alpha = expf(v - m[d * H + h]) / den[d * H + h];
  const float* fsrc = feat + ((size_t)s * H + h) * D;
  float* odst = out + ((size_t)d * H + h) * D;
  for (int i = lane; i < D; i += 32) {
    atomicAdd(&odst[i], fsrc[i] * alpha);
  }
}

// ---------------------------------------------------------------------------
// In-place bias + relu (layer-1 epilogue). cmask = C-1, C power of two.
// ---------------------------------------------------------------------------
__global__ void gat_bias_relu(float* __restrict__ p,
                              const float* __restrict__ bias,
                              size_t n, int cmask) {
  size_t i = (size_t)blockIdx.x * blockDim.x + threadIdx.x;
  if (i < n) {
    float v = p[i] + bias[i & (size_t)cmask];
    p[i] = v > 0.0f ? v : 0.0f;
  }
}

// ---------------------------------------------------------------------------
// Finalize: graph_output = relu(acc + bias2) (in place in d_out slice),
// graph_embedding[node] = sum over the 64 output dims. One wave per node.
// ---------------------------------------------------------------------------
__global__ void gat_finalize(float* __restrict__ gout,
                             const float* __restrict__ bias2,
                             float* __restrict__ emb, int N, int D) {
  int wave = (int)(((size_t)blockIdx.x * blockDim.x + threadIdx.x) >> 5);
  int lane = threadIdx.x & 31;
  if (wave >= N) return;
  float* p = gout + (size_t)wave * D;
  float s = 0.0f;
  for (int i = lane; i < D; i += 32) {
    float v = p[i] + bias2[i];
    v = v > 0.0f ? v : 0.0f;
    p[i] = v;
    s += v;
  }
#pragma unroll
  for (int o = 16; o > 0; o >>= 1) s += __shfl_xor(s, o, 32);
  if (lane == 0) emb[wave] = s;
}

// ---------------------------------------------------------------------------
// Host-side launch sequence.
// ---------------------------------------------------------------------------
static inline int ceil_div(long long a, long long b) { return (int)((a + b - 1) / b); }

extern "C" void kernel_launch(void* const* d_in, const int* in_sizes, int n_in,
                              void* d_out, int out_size, void* d_ws, size_t ws_size,
                              hipStream_t stream) {
  const float* x       = (const float*)d_in[0];
  const int*   src     = (const int*)d_in[1];
  const int*   dst     = (const int*)d_in[2];
  const float* W1      = (const float*)d_in[3];
  const float* attn_l1 = (const float*)d_in[4];
  const float* attn_r1 = (const float*)d_in[5];
  const float* bias1   = (const float*)d_in[6];
  const float* W2      = (const float*)d_in[7];
  const float* attn_l2 = (const float*)d_in[8];
  const float* attn_r2 = (const float*)d_in[9];
  const float* bias2   = (const float*)d_in[10];

  const int C1 = 256, H1 = 4, D1 = 64;   // layer 1: 4 heads x 64
  const int C2 = 64,  H2 = 1, D2 = 64;   // layer 2: 1 head  x 64
  const int F1 = in_sizes[3] / C1;       // 128
  const int N  = in_sizes[0] / F1;       // 50000
  const int E  = in_sizes[1];            // 800000

  // Workspace layout (floats).
  float* ws    = (float*)d_ws;
  float* feat1 = ws;                               // N*C1
  float* agg1  = feat1 + (size_t)N * C1;           // N*C1 (becomes h after relu)
  float* feat2 = agg1 + (size_t)N * C1;            // N*C2
  float* el1   = feat2 + (size_t)N * C2;           // N*H1
  float* er1   = el1 + (size_t)N * H1;             // N*H1
  float* m1    = er1 + (size_t)N * H1;             // N*H1
  float* den1  = m1 + (size_t)N * H1;              // N*H1
  float* el2   = den1 + (size_t)N * H1;            // N
  float* er2   = el2 + (size_t)N;                  // N
  float* m2    = er2 + (size_t)N;                  // N
  float* den2  = m2 + (size_t)N;                   // N

  // Output layout: [embedding (N)] then [graph_output (N*64)].
  float* emb  = (float*)d_out;
  float* gout = emb + (size_t)N;

  const int TB = 256; // 8 waves/block

  // ---- Layer 1 ----
  {
    int waves = ceil_div(N, 16) * (C1 / 16);
    gat_wmma_gemm_f32<<<ceil_div((long long)waves * 32, TB), TB, 0, stream>>>(
        x, W1, feat1, N, F1, C1);
  }
  gat_attn_scores<<<ceil_div((long long)N * H1 * 32, TB), TB, 0, stream>>>(
      feat1, attn_l1, attn_r1, el1, er1, N * H1, H1, D1);

  gat_fill_kernel<<<ceil_div((long long)N * H1, TB), TB, 0, stream>>>(m1, -INFINITY, (size_t)N * H1);
  gat_fill_kernel<<<ceil_div((long long)N * H1, TB), TB, 0, stream>>>(den1, 0.0f, (size_t)N * H1);
  gat_fill_kernel<<<ceil_div((long long)N * C1, TB), TB, 0, stream>>>(agg1, 0.0f, (size_t)N * C1);

  gat_edge_max<<<ceil_div((long long)E * H1, TB), TB, 0, stream>>>(el1, er1, src, dst, m1, E, H1);
  gat_edge_expsum<<<ceil_div((long long)E * H1, TB), TB, 0, stream>>>(el1, er1, m1, den1, src, dst, E, H1);
  gat_aggregate<<<ceil_div((long long)E * H1 * 32, TB), TB, 0, stream>>>(
      feat1, el1, er1, m1, den1, src, dst, agg1, E, H1, D1);

  gat_bias_relu<<<ceil_div((long long)N * C1, TB), TB, 0, stream>>>(
      agg1, bias1, (size_t)N * C1, C1 - 1);

  // ---- Layer 2 ----
  {
    int waves = ceil_div(N, 16) * (C2 / 16);
    gat_wmma_gemm_f32<<<ceil_div((long long)waves * 32, TB), TB, 0, stream>>>(
        agg1, W2, feat2, N, C1, C2);
  }
  gat_attn_scores<<<ceil_div((long long)N * H2 * 32, TB), TB, 0, stream>>>(
      feat2, attn_l2, attn_r2, el2, er2, N * H2, H2, D2);

  gat_fill_kernel<<<ceil_div((long long)N, TB), TB, 0, stream>>>(m2, -INFINITY, (size_t)N);
  gat_fill_kernel<<<ceil_div((long long)N, TB), TB, 0, stream>>>(den2, 0.0f, (size_t)N);
  gat_fill_kernel<<<ceil_div((long long)N * C2, TB), TB, 0, stream>>>(gout, 0.0f, (size_t)N * C2);

  gat_edge_max<<<ceil_div((long long)E * H2, TB), TB, 0, stream>>>(el2, er2, src, dst, m2, E, H2);
  gat_edge_expsum<<<ceil_div((long long)E * H2, TB), TB, 0, stream>>>(el2, er2, m2, den2, src, dst, E, H2);
  gat_aggregate<<<ceil_div((long long)E * H2 * 32, TB), TB, 0, stream>>>(
      feat2, el2, er2, m2, den2, src, dst, gout, E, H2, D2);

  gat_finalize<<<ceil_div((long long)N * 32, TB), TB, 0, stream>>>(gout, bias2, emb, N, C2);
}